// AUVLSTMDeltaV_60979945668885
// MI455X (gfx1250) — compile-verified
//
#include <hip/hip_runtime.h>

#define NLAYERS 5
#define TSTEPS  32
#define NEGS    0.1f
#define BNEPS   1e-5f

typedef float v2f __attribute__((ext_vector_type(2)));
typedef float v8f __attribute__((ext_vector_type(8)));

__device__ __forceinline__ float sigm(float x) {
    return __builtin_amdgcn_rcpf(1.0f + __expf(-x));
}
__device__ __forceinline__ float tanh_fast(float x) {
    return 2.0f * __builtin_amdgcn_rcpf(1.0f + __expf(-2.0f * x)) - 1.0f;
}
__device__ __forceinline__ float lrelu(float x) {
    return x > 0.0f ? x : NEGS * x;
}

// ---------------------------------------------------------------------------
// K1: per-sample 5-layer LSTM recurrence (H=1), memory-bandwidth bound.
// One lane = one sample. Inputs buffered 4 timesteps at a time with aligned
// float4 (global_load_b128) reads. Writes y (=top-layer h at t=0) to ws,
// hT/cT to d_out, and per-block partial sums of y / y^2 for BN1 stats.
// ---------------------------------------------------------------------------
__global__ __launch_bounds__(256) void k1_lstm(
    const float* __restrict__ r, const float* __restrict__ v,
    const float* __restrict__ u,
    const float* __restrict__ w_ih0,      // [4][21]
    const float* __restrict__ w_ih_rest,  // [4][4][1]
    const float* __restrict__ w_hh,       // [5][4][1]
    float* __restrict__ out, int k,
    float* __restrict__ ws_y, float* __restrict__ part1)
{
    const int i = blockIdx.x * blockDim.x + threadIdx.x;

    // Uniform weights -> scalar registers
    float Wih0[4][21];
#pragma unroll
    for (int g = 0; g < 4; ++g)
#pragma unroll
        for (int j = 0; j < 21; ++j) Wih0[g][j] = w_ih0[g * 21 + j];
    float Wr[4][4], Wh[5][4];
#pragma unroll
    for (int l = 0; l < 4; ++l)
#pragma unroll
        for (int g = 0; g < 4; ++g) Wr[l][g] = w_ih_rest[l * 4 + g];
#pragma unroll
    for (int l = 0; l < 5; ++l)
#pragma unroll
        for (int g = 0; g < 4; ++g) Wh[l][g] = w_hh[l * 4 + g];

    float h[NLAYERS] = {0.f, 0.f, 0.f, 0.f, 0.f};
    float c[NLAYERS] = {0.f, 0.f, 0.f, 0.f, 0.f};
    float y0 = 0.f;

    // Per-sample contiguous blocks, all 16B aligned:
    // r: 32*9 floats (1152B), v/u: 32*6 floats (768B)
    const float4* rp = (const float4*)(r + (size_t)i * TSTEPS * 9);
    const float4* vp = (const float4*)(v + (size_t)i * TSTEPS * 6);
    const float4* up = (const float4*)(u + (size_t)i * TSTEPS * 6);

#pragma unroll 1
    for (int tg = 0; tg < 8; ++tg) {          // 4 timesteps per group
        if (tg + 1 < 8) {                      // software prefetch next group
            __builtin_prefetch(rp + (tg + 1) * 9, 0, 0);
            __builtin_prefetch(vp + (tg + 1) * 6, 0, 0);
            __builtin_prefetch(up + (tg + 1) * 6, 0, 0);
        }
        float xr[36], xv[24], xu[24];
#pragma unroll
        for (int q = 0; q < 9; ++q) {
            float4 t4 = rp[tg * 9 + q];
            xr[q*4+0] = t4.x; xr[q*4+1] = t4.y; xr[q*4+2] = t4.z; xr[q*4+3] = t4.w;
        }
#pragma unroll
        for (int q = 0; q < 6; ++q) {
            float4 t4 = vp[tg * 6 + q];
            xv[q*4+0] = t4.x; xv[q*4+1] = t4.y; xv[q*4+2] = t4.z; xv[q*4+3] = t4.w;
        }
#pragma unroll
        for (int q = 0; q < 6; ++q) {
            float4 t4 = up[tg * 6 + q];
            xu[q*4+0] = t4.x; xu[q*4+1] = t4.y; xu[q*4+2] = t4.z; xu[q*4+3] = t4.w;
        }
#pragma unroll
        for (int tt = 0; tt < 4; ++tt) {
            float x[21];
#pragma unroll
            for (int j = 0; j < 9; ++j) x[j] = xr[tt * 9 + j];
#pragma unroll
            for (int j = 0; j < 6; ++j) x[9 + j] = xv[tt * 6 + j];
#pragma unroll
            for (int j = 0; j < 6; ++j) x[15 + j] = xu[tt * 6 + j];

            // Layer 0: 4x21 matvec + recurrent term
            float g4[4];
#pragma unroll
            for (int g = 0; g < 4; ++g) {
                float acc = Wh[0][g] * h[0];
#pragma unroll
                for (int j = 0; j < 21; ++j) acc = fmaf(Wih0[g][j], x[j], acc);
                g4[g] = acc;
            }
            {
                float ig = sigm(g4[0]), fg = sigm(g4[1]);
                float gg = tanh_fast(g4[2]), og = sigm(g4[3]);
                c[0] = fg * c[0] + ig * gg;
                h[0] = og * tanh_fast(c[0]);
            }
            // Layers 1..4 (scalar in/out)
#pragma unroll
            for (int l = 1; l < NLAYERS; ++l) {
                float ig = sigm(Wr[l-1][0] * h[l-1] + Wh[l][0] * h[l]);
                float fg = sigm(Wr[l-1][1] * h[l-1] + Wh[l][1] * h[l]);
                float gg = tanh_fast(Wr[l-1][2] * h[l-1] + Wh[l][2] * h[l]);
                float og = sigm(Wr[l-1][3] * h[l-1] + Wh[l][3] * h[l]);
                c[l] = fg * c[l] + ig * gg;
                h[l] = og * tanh_fast(c[l]);
            }
            if (tg == 0 && tt == 0) y0 = h[NLAYERS - 1];  // outs[0]
        }
    }

    ws_y[i] = y0;
    const size_t dvLen = (size_t)k * 6;
#pragma unroll
    for (int l = 0; l < NLAYERS; ++l) {
        out[dvLen + (size_t)l * k + i] = h[l];
        out[dvLen + (size_t)NLAYERS * k + (size_t)l * k + i] = c[l];
    }

    // Deterministic per-block reduction of sum(y), sum(y^2)
    __shared__ float sacc[16];
    float s = y0, s2 = y0 * y0;
#pragma unroll
    for (int off = 16; off > 0; off >>= 1) {
        s  += __shfl_xor(s,  off, 32);
        s2 += __shfl_xor(s2, off, 32);
    }
    const int wave = threadIdx.x >> 5, lane = threadIdx.x & 31;
    if (lane == 0) { sacc[wave * 2] = s; sacc[wave * 2 + 1] = s2; }
    __syncthreads();
    if (threadIdx.x == 0) {
        float ts = 0.f, ts2 = 0.f;
        for (int w = 0; w < 8; ++w) { ts += sacc[w * 2]; ts2 += sacc[w * 2 + 1]; }
        part1[blockIdx.x * 2]     = ts;
        part1[blockIdx.x * 2 + 1] = ts2;
    }
}

// K1b: finalize BN1 batch stats (mean, biased var of y)
__global__ void k1b_reduce(const float* __restrict__ part1, int nblk, float invk,
                           float* __restrict__ stats1)
{
    if (threadIdx.x == 0) {
        float s = 0.f, s2 = 0.f;
        for (int b = 0; b < nblk; ++b) { s += part1[2 * b]; s2 += part1[2 * b + 1]; }
        float m = s * invk;
        stats1[0] = m;
        stats1[1] = s2 * invk - m * m;
    }
}

// ---------------------------------------------------------------------------
// Shared head helpers
// ---------------------------------------------------------------------------
__device__ __forceinline__ void compute_y1(
    float y, float m_y, float v_y,
    const float* __restrict__ W1, const float* __restrict__ g1,
    const float* __restrict__ b1, float y1[32])
{
#pragma unroll
    for (int j = 0; j < 32; ++j) {
        float w  = W1[j];                 // W1 is [32,1]
        float z  = w * y;
        float mj = w * m_y;
        float vj = w * w * v_y;
        float n  = g1[j] * (z - mj) * __builtin_amdgcn_rsqf(vj + BNEPS) + b1[j];
        y1[j] = lrelu(n);
    }
}

// z2 = y1 @ W2^T for 32 samples per wave via V_WMMA_F32_16X16X4_F32.
// lw2: LDS copy of W2 [n*32+k]. lbuf: per-wave 32x32 LDS staging tile.
// A 16x4 f32 layout: lanes 0-15 = rows M, K = vgpr + 2*(lane>=16)
// B 4x16  f32 layout: lanes 0-15 = cols N, K row = vgpr + 2*(lane>=16)
// D 16x16 f32 layout: VGPR j -> M = j + 8*(lane>=16), N = lane&15
__device__ __forceinline__ void wmma_z2(
    const float y1[32], const float* __restrict__ lw2,
    float* __restrict__ lbuf, float z2[32])
{
    const int lane = threadIdx.x & 31;
    const int half = lane >> 4;
    const int l16  = lane & 15;

    // stage this lane's y1 row (sample row = lane within wave tile)
#pragma unroll
    for (int j = 0; j < 32; ++j) lbuf[lane * 32 + j] = y1[j];
    __syncthreads();

    v8f acc[2][2] = {};   // [m-tile][n-tile]
#pragma unroll
    for (int kk = 0; kk < 8; ++kk) {
        v2f b0, b1v;
        b0.x  = lw2[(0 * 16 + l16) * 32 + kk * 4 + half * 2 + 0];
        b0.y  = lw2[(0 * 16 + l16) * 32 + kk * 4 + half * 2 + 1];
        b1v.x = lw2[(1 * 16 + l16) * 32 + kk * 4 + half * 2 + 0];
        b1v.y = lw2[(1 * 16 + l16) * 32 + kk * 4 + half * 2 + 1];
#pragma unroll
        for (int mt = 0; mt < 2; ++mt) {
            v2f a;
            a.x = lbuf[(mt * 16 + l16) * 32 + kk * 4 + half * 2 + 0];
            a.y = lbuf[(mt * 16 + l16) * 32 + kk * 4 + half * 2 + 1];
            acc[mt][0] = __builtin_amdgcn_wmma_f32_16x16x4_f32(
                false, a, false, b0,  (short)0, acc[mt][0], false, false);
            acc[mt][1] = __builtin_amdgcn_wmma_f32_16x16x4_f32(
                false, a, false, b1v, (short)0, acc[mt][1], false, false);
        }
    }
    __syncthreads();   // A reads done; safe to overwrite lbuf with D
#pragma unroll
    for (int mt = 0; mt < 2; ++mt)
#pragma unroll
        for (int nt = 0; nt < 2; ++nt)
#pragma unroll
            for (int j = 0; j < 8; ++j)
                lbuf[(mt * 16 + half * 8 + j) * 32 + nt * 16 + l16] = acc[mt][nt][j];
    __syncthreads();
#pragma unroll
    for (int j = 0; j < 32; ++j) z2[j] = lbuf[lane * 32 + j];
}

// ---------------------------------------------------------------------------
// K2: per-feature batch stats of z2 (BN2, training mode), deterministic
// ---------------------------------------------------------------------------
__global__ __launch_bounds__(256) void k2_stats(
    const float* __restrict__ ws_y, const float* __restrict__ stats1,
    const float* __restrict__ W1, const float* __restrict__ g1,
    const float* __restrict__ b1, const float* __restrict__ W2,
    float* __restrict__ part2)
{
    __shared__ float lw2[1024];
    __shared__ float lbuf[8 * 1024];
    __shared__ float waveacc[8][64];
    const int tid = threadIdx.x;
    for (int q = tid; q < 1024; q += 256) lw2[q] = W2[q];
    __syncthreads();

    const int i = blockIdx.x * 256 + tid;
    float y1[32];
    compute_y1(ws_y[i], stats1[0], stats1[1], W1, g1, b1, y1);
    float z2[32];
    wmma_z2(y1, lw2, lbuf + (tid >> 5) * 1024, z2);

    const int lane = tid & 31, wave = tid >> 5;
#pragma unroll
    for (int j = 0; j < 32; ++j) {
        float s = z2[j], s2 = z2[j] * z2[j];
#pragma unroll
        for (int off = 16; off > 0; off >>= 1) {
            s  += __shfl_xor(s,  off, 32);
            s2 += __shfl_xor(s2, off, 32);
        }
        if (lane == 0) { waveacc[wave][j] = s; waveacc[wave][32 + j] = s2; }
    }
    __syncthreads();
    if (tid < 64) {
        float t = 0.f;
        for (int w = 0; w < 8; ++w) t += waveacc[w][tid];
        part2[blockIdx.x * 64 + tid] = t;
    }
}

// K2b: finalize BN2 stats: stats2[0..31]=mean, stats2[32..63]=var
__global__ void k2b_reduce(const float* __restrict__ part2, int nblk, float invk,
                           float* __restrict__ stats2)
{
    const int j = threadIdx.x;
    if (j < 32) {
        float s = 0.f, s2 = 0.f;
        for (int b = 0; b < nblk; ++b) {
            s  += part2[b * 64 + j];
            s2 += part2[b * 64 + 32 + j];
        }
        float m = s * invk;
        stats2[j]      = m;
        stats2[32 + j] = s2 * invk - m * m;
    }
}

// ---------------------------------------------------------------------------
// K3: full head: y1 -> WMMA z2 -> BN2 -> LReLU -> W3 -> dv
// ---------------------------------------------------------------------------
__global__ __launch_bounds__(256) void k3_head(
    const float* __restrict__ ws_y, const float* __restrict__ stats1,
    const float* __restrict__ stats2,
    const float* __restrict__ W1, const float* __restrict__ g1,
    const float* __restrict__ b1, const float* __restrict__ g2,
    const float* __restrict__ b2, const float* __restrict__ W2,
    const float* __restrict__ W3, float* __restrict__ out)
{
    __shared__ float lw2[1024];
    __shared__ float lbuf[8 * 1024];
    const int tid = threadIdx.x;
    for (int q = tid; q < 1024; q += 256) lw2[q] = W2[q];
    __syncthreads();

    const int i = blockIdx.x * 256 + tid;
    float y1[32];
    compute_y1(ws_y[i], stats1[0], stats1[1], W1, g1, b1, y1);
    float z2[32];
    wmma_z2(y1, lw2, lbuf + (tid >> 5) * 1024, z2);

    float y2[32];
#pragma unroll
    for (int j = 0; j < 32; ++j) {
        float n = g2[j] * (z2[j] - stats2[j]) *
                  __builtin_amdgcn_rsqf(stats2[32 + j] + BNEPS) + b2[j];
        y2[j] = lrelu(n);
    }
#pragma unroll
    for (int n = 0; n < 6; ++n) {
        float acc = 0.f;
#pragma unroll
        for (int j = 0; j < 32; ++j) acc = fmaf(W3[n * 32 + j], y2[j], acc);
        out[(size_t)i * 6 + n] = acc;
    }
}

// ---------------------------------------------------------------------------
extern "C" void kernel_launch(void* const* d_in, const int* in_sizes, int n_in,
                              void* d_out, int out_size, void* d_ws, size_t ws_size,
                              hipStream_t stream)
{
    const float* r         = (const float*)d_in[0];
    const float* v         = (const float*)d_in[1];
    const float* u         = (const float*)d_in[2];
    const float* w_ih0     = (const float*)d_in[3];
    const float* w_ih_rest = (const float*)d_in[4];
    const float* w_hh      = (const float*)d_in[5];
    const float* W1        = (const float*)d_in[6];
    const float* g1        = (const float*)d_in[7];
    const float* b1        = (const float*)d_in[8];
    const float* W2        = (const float*)d_in[9];
    const float* g2        = (const float*)d_in[10];
    const float* b2        = (const float*)d_in[11];
    const float* W3        = (const float*)d_in[12];

    const int k    = in_sizes[0] / (TSTEPS * 9);   // 65536
    const int nblk = k / 256;                      // 256; k is a multiple of 256
    const float invk = 1.0f / (float)k;

    float* ws     = (float*)d_ws;
    float* ws_y   = ws;                    // [k]
    float* part1  = ws_y + k;              // [2*nblk]
    float* stats1 = part1 + 2 * nblk;      // [2]
    float* part2  = stats1 + 2;            // [64*nblk]
    float* stats2 = part2 + 64 * nblk;     // [64]
    float* out    = (float*)d_out;

    k1_lstm   <<<nblk, 256, 0, stream>>>(r, v, u, w_ih0, w_ih_rest, w_hh,
                                         out, k, ws_y, part1);
    k1b_reduce<<<1,    64,  0, stream>>>(part1, nblk, invk, stats1);
    k2_stats  <<<nblk, 256, 0, stream>>>(ws_y, stats1, W1, g1, b1, W2, part2);
    k2b_reduce<<<1,    32,  0, stream>>>(part2, nblk, invk, stats2);
    k3_head   <<<nblk, 256, 0, stream>>>(ws_y, stats1, stats2, W1, g1, b1,
                                         g2, b2, W2, W3, out);
}